// L1_58428735095539
// MI455X (gfx1250) — compile-verified
//
#include <hip/hip_runtime.h>

// Negative pairwise L1 distance: out[i,j] = -sum_d |x[i,d] - y[j,d]|
// x: [N,D], y: [M,D], out: [N,M]; N=M=2048, D=128, fp32.
// VALU-bound problem (no bilinear form -> no WMMA mapping). CDNA5 paths used:
// async global->LDS staging (ASYNCcnt) + wave32 tiling + bank-conflict-free
// d-major LDS layout read with ds_load_b128.

constexpr int Nn = 2048;
constexpr int Mm = 2048;
constexpr int Dd = 128;
constexpr int BM = 64;   // x-rows per block tile
constexpr int BN = 64;   // y-rows per block tile

#if defined(__has_builtin)
#if __has_builtin(__builtin_amdgcn_global_load_async_to_lds_b32)
#define L1_HAVE_ASYNC_LDS 1
#endif
#endif

#if defined(L1_HAVE_ASYNC_LDS)
typedef __attribute__((address_space(1))) int g_i32;  // global dword
typedef __attribute__((address_space(3))) int l_i32;  // LDS dword
#endif

__global__ __launch_bounds__(256) void l1neg_tile_kernel(
    const float* __restrict__ x, const float* __restrict__ y,
    float* __restrict__ out) {
  // d-major tiles: xs[d*BM + i], ys[d*BN + j]  (32 KB each)
  __shared__ float xs[Dd * BM];
  __shared__ float ys[Dd * BN];

  const int tid = threadIdx.x;          // 0..255
  const int i0  = blockIdx.x * BM;      // output row base
  const int j0  = blockIdx.y * BN;      // output col base

  // ---------------- stage tiles into LDS (transposed to d-major) -----------
#if defined(L1_HAVE_ASYNC_LDS)
  {
    // 256 threads; thread handles row r = tid&63, d-chunk (tid>>6)*32.
    // Per async op: one dword; per-lane LDS scatter performs the transpose.
    const int r  = tid & 63;
    const int dc = (tid >> 6) * 32;
    g_i32* gx = (g_i32*)(x + (size_t)(i0 + r) * Dd + dc);
    g_i32* gy = (g_i32*)(y + (size_t)(j0 + r) * Dd + dc);
    l_i32* lx = (l_i32*)&xs[dc * BM + r];
    l_i32* ly = (l_i32*)&ys[dc * BN + r];
#pragma unroll
    for (int u = 0; u < 32; ++u) {
      __builtin_amdgcn_global_load_async_to_lds_b32(gx + u, lx + (size_t)u * BM, 0, 0);
      __builtin_amdgcn_global_load_async_to_lds_b32(gy + u, ly + (size_t)u * BN, 0, 0);
    }
  }
#if defined(__has_builtin) && __has_builtin(__builtin_amdgcn_s_wait_asynccnt)
  __builtin_amdgcn_s_wait_asynccnt(0);
#else
  asm volatile("s_wait_asynccnt 0" ::: "memory");
#endif
#else
  // Fallback: coalesced float4 global loads + 4 scalar LDS stores (transpose).
  {
    constexpr int Q = (BM * Dd) / 4;  // 2048 float4 slots per tile
    for (int k = tid; k < Q; k += 256) {
      const int r  = k / (Dd / 4);        // tile row
      const int dq = (k % (Dd / 4)) * 4;  // d base
      const float4 xv = *(const float4*)(x + (size_t)(i0 + r) * Dd + dq);
      xs[(dq + 0) * BM + r] = xv.x;
      xs[(dq + 1) * BM + r] = xv.y;
      xs[(dq + 2) * BM + r] = xv.z;
      xs[(dq + 3) * BM + r] = xv.w;
      const float4 yv = *(const float4*)(y + (size_t)(j0 + r) * Dd + dq);
      ys[(dq + 0) * BN + r] = yv.x;
      ys[(dq + 1) * BN + r] = yv.y;
      ys[(dq + 2) * BN + r] = yv.z;
      ys[(dq + 3) * BN + r] = yv.w;
    }
  }
#endif
  __syncthreads();

  // ---------------- compute 4x4 register sub-tile per thread ---------------
  const int ix = (tid & 15) * 4;   // 4 x-rows of this thread
  const int jy = (tid >> 4) * 4;   // 4 y-rows of this thread

  float acc[4][4];
#pragma unroll
  for (int a = 0; a < 4; ++a)
#pragma unroll
    for (int b = 0; b < 4; ++b) acc[a][b] = 0.0f;

#pragma unroll 4
  for (int d = 0; d < Dd; ++d) {
    const float4 xv = *(const float4*)&xs[d * BM + ix];  // ds_load_b128
    const float4 yv = *(const float4*)&ys[d * BN + jy];  // ds_load_b128
    const float xa[4] = {xv.x, xv.y, xv.z, xv.w};
    const float ya[4] = {yv.x, yv.y, yv.z, yv.w};
#pragma unroll
    for (int a = 0; a < 4; ++a)
#pragma unroll
      for (int b = 0; b < 4; ++b)
        acc[a][b] += __builtin_fabsf(xa[a] - ya[b]);  // v_sub + v_add |abs|
  }

  // ---------------- write 4 rows x float4 ----------------------------------
#pragma unroll
  for (int a = 0; a < 4; ++a) {
    float4 o;
    o.x = -acc[a][0];
    o.y = -acc[a][1];
    o.z = -acc[a][2];
    o.w = -acc[a][3];
    *(float4*)(out + (size_t)(i0 + ix + a) * Mm + (j0 + jy)) = o;
  }
}

extern "C" void kernel_launch(void* const* d_in, const int* in_sizes, int n_in,
                              void* d_out, int out_size, void* d_ws, size_t ws_size,
                              hipStream_t stream) {
  (void)in_sizes; (void)n_in; (void)out_size; (void)d_ws; (void)ws_size;
  const float* x = (const float*)d_in[0];
  const float* y = (const float*)d_in[1];
  float* out = (float*)d_out;

  dim3 grid(Nn / BM, Mm / BN);  // 32 x 32
  dim3 block(256);
  l1neg_tile_kernel<<<grid, block, 0, stream>>>(x, y, out);
}